// ExpertRouter_32238024524298
// MI455X (gfx1250) — compile-verified
//
#include <hip/hip_runtime.h>
#include <hip/hip_bf16.h>

// ============================================================================
// MoE Expert Router for MI455X (gfx1250, wave32)
//
// router_logits = x[8192,4096] @ W[64,4096]^T   -> f32 WMMA 16x16x4
// + noisy top-2, softmax weights, expert counts, mean router prob, LB loss.
//
// Roofline: x stream = 128 MB -> ~5.6 us floor @ 23.3 TB/s; 4.3 GFLOP f32 is
// far below any WMMA ceiling, so native f32 WMMA (no down-conversion) is
// correct. GEMM is mapped so every x byte is read from HBM exactly once:
// each block owns one 16-row M-tile; its 4 waves split the K dimension
// (1024 each) and each wave computes ALL 4 expert tiles, so one x float4
// load feeds 8 WMMAs. Partials are reduced through 16 KB of LDS.
//
// d_out layout (flat f32, reference tuple order):
//   [0      , 16384) top_k_indices (as float) [8192][2]
//   [16384  , 32768) expert_weights           [8192][2]
//   [32768  ,557056) router_logits            [8192][64]
//   [557056]         load_balance_loss
// d_ws: [0..63] expert counts, [64..127] softmax column sums.
// ============================================================================

#define BATCH 8192
#define DIM   4096
#define NE    64
#define KCHUNK (DIM / 4)   // K per wave

typedef __attribute__((ext_vector_type(2))) float v2f;
typedef __attribute__((ext_vector_type(8))) float v8f;

// ---------------------------------------------------------------------------
// Zero the 128-float workspace (must run every launch: graph replays).
// ---------------------------------------------------------------------------
__global__ void router_init(float* __restrict__ ws) {
    ws[threadIdx.x] = 0.0f;
}

// ---------------------------------------------------------------------------
// GEMM: logits[m][e] = sum_k x[m][k] * W[e][k]
// Grid: 512 blocks (M tiles of 16). Block: 128 threads = 4 waves.
// Wave w covers K in [w*1024, (w+1)*1024) and accumulates 4 expert tiles.
// Per K-step of 8: 1 x float4 + 4 W float4 loads -> 8 v_wmma_f32_16x16x4_f32.
// A and B use the same per-lane physical-K->slot mapping, so the slot
// permutation cancels in the accumulated dot product (sum over K slots).
// Epilogue: LDS reduction of the 4 per-wave partials, then one store.
// ---------------------------------------------------------------------------
__global__ __launch_bounds__(128) void router_gemm(const float* __restrict__ x,
                                                   const float* __restrict__ W,
                                                   float* __restrict__ logits) {
    __shared__ float s_red[4 * 4 * 32 * 8];      // [wave][etile][lane][8] = 16 KB

    const int wave   = threadIdx.x >> 5;         // 0..3 -> K chunk
    const int lane   = threadIdx.x & 31;
    const int r      = lane & 15;                // M-in-tile (A) / N-in-tile (B)
    const int h      = lane >> 4;                // half-wave: K-slot group 0/1
    const int m_base = blockIdx.x * 16;
    const int kbeg   = wave * KCHUNK;

    const float* __restrict__ xp = x + (size_t)(m_base + r) * DIM + kbeg + 4 * h;
    const float* __restrict__ wp = W + (size_t)r * DIM + kbeg + 4 * h;

    // W is 1 MB total: L2-resident; hint it in.
    __builtin_prefetch(wp, 0, 3);

    v8f acc[4] = {};                             // 4 expert tiles, 8 VGPRs each

    for (int k = 0; k < KCHUNK; k += 8) {
        const float4 xa = *(const float4*)(xp + k);
        const v2f a0 = { xa.x, xa.y };
        const v2f a1 = { xa.z, xa.w };
#pragma unroll
        for (int t = 0; t < 4; ++t) {
            const float4 wb = *(const float4*)(wp + (size_t)(16 * t) * DIM + k);
            const v2f b0 = { wb.x, wb.y };
            const v2f b1 = { wb.z, wb.w };
            acc[t] = __builtin_amdgcn_wmma_f32_16x16x4_f32(
                /*neg_a=*/false, a0, /*neg_b=*/false, b0,
                /*c_mod=*/(short)0, acc[t], /*reuse_a=*/false, /*reuse_b=*/false);
            acc[t] = __builtin_amdgcn_wmma_f32_16x16x4_f32(
                false, a1, false, b1, (short)0, acc[t], false, false);
        }
    }

    // Dump partials to LDS: s_red[wave][t][lane][0..7]
#pragma unroll
    for (int t = 0; t < 4; ++t) {
        float* dst = &s_red[((wave * 4 + t) * 32 + lane) * 8];
        *(float4*)(dst + 0) = make_float4(acc[t][0], acc[t][1], acc[t][2], acc[t][3]);
        *(float4*)(dst + 4) = make_float4(acc[t][4], acc[t][5], acc[t][6], acc[t][7]);
    }
    __syncthreads();

    // Reduce across the 4 waves. Thread (t = tid>>5, l = tid&31) owns the
    // 8 accumulator elements of (etile t, lane l).
    {
        const int t = threadIdx.x >> 5;
        const int l = threadIdx.x & 31;
        float4 lo = make_float4(0.f, 0.f, 0.f, 0.f);
        float4 hi = make_float4(0.f, 0.f, 0.f, 0.f);
#pragma unroll
        for (int w = 0; w < 4; ++w) {
            const float* src = &s_red[((w * 4 + t) * 32 + l) * 8];
            const float4 a = *(const float4*)(src + 0);
            const float4 b = *(const float4*)(src + 4);
            lo.x += a.x; lo.y += a.y; lo.z += a.z; lo.w += a.w;
            hi.x += b.x; hi.y += b.y; hi.z += b.z; hi.w += b.w;
        }
        // D layout: element i of (t,l) -> row = i + 8*(l>>4), col = 16*t + (l&15)
        const int col = 16 * t + (l & 15);
        const int rbase = m_base + 8 * (l >> 4);
        logits[(size_t)(rbase + 0) * NE + col] = lo.x;
        logits[(size_t)(rbase + 1) * NE + col] = lo.y;
        logits[(size_t)(rbase + 2) * NE + col] = lo.z;
        logits[(size_t)(rbase + 3) * NE + col] = lo.w;
        logits[(size_t)(rbase + 4) * NE + col] = hi.x;
        logits[(size_t)(rbase + 5) * NE + col] = hi.y;
        logits[(size_t)(rbase + 6) * NE + col] = hi.z;
        logits[(size_t)(rbase + 7) * NE + col] = hi.w;
    }
}

// ---------------------------------------------------------------------------
// Per-row epilogue: one wave32 per row, 2 experts per lane.
// noisy top-2 (+softmax weights), clean-row softmax, per-expert count &
// prob sums accumulated through LDS then one global atomic per expert/block.
// ---------------------------------------------------------------------------
__global__ __launch_bounds__(256) void router_rowwise(
    const float* __restrict__ logits, const float* __restrict__ noise,
    float* __restrict__ out_idx, float* __restrict__ out_w,
    float* __restrict__ g_counts, float* __restrict__ g_psum) {
    __shared__ float s_counts[NE];
    __shared__ float s_psum[NE];
    if (threadIdx.x < NE) { s_counts[threadIdx.x] = 0.0f; s_psum[threadIdx.x] = 0.0f; }
    __syncthreads();

    const int lane = threadIdx.x & 31;
    const int row  = blockIdx.x * 8 + (threadIdx.x >> 5);

    const float l0 = logits[row * NE + lane];
    const float l1 = logits[row * NE + lane + 32];
    const float z0 = l0 + 0.1f * noise[row * NE + lane];
    const float z1 = l1 + 0.1f * noise[row * NE + lane + 32];

    // ---- warp top-2 over 64 noisy logits ----
    float v1, v2; int i1, i2;
    if (z0 >= z1) { v1 = z0; i1 = lane;      v2 = z1; i2 = lane + 32; }
    else          { v1 = z1; i1 = lane + 32; v2 = z0; i2 = lane;      }
    for (int off = 16; off > 0; off >>= 1) {
        float ov1 = __shfl_down(v1, off, 32);
        float ov2 = __shfl_down(v2, off, 32);
        int   oi1 = __shfl_down(i1, off, 32);
        int   oi2 = __shfl_down(i2, off, 32);
        if (ov1 > v1) {
            if (v1 > ov2) { v2 = v1; i2 = i1; } else { v2 = ov2; i2 = oi2; }
            v1 = ov1; i1 = oi1;
        } else if (ov1 > v2) {
            v2 = ov1; i2 = oi1;
        }
    }
    if (lane == 0) {
        const float a = __expf(v2 - v1);             // softmax over 2 vals
        const float inv = 1.0f / (1.0f + a);
        out_idx[row * 2 + 0] = (float)i1;
        out_idx[row * 2 + 1] = (float)i2;
        out_w[row * 2 + 0]   = inv;
        out_w[row * 2 + 1]   = a * inv;
        atomicAdd(&s_counts[i1], 1.0f);
        atomicAdd(&s_counts[i2], 1.0f);
    }

    // ---- clean-logit row softmax, accumulate column sums ----
    float m = fmaxf(l0, l1);
    for (int off = 16; off > 0; off >>= 1) m = fmaxf(m, __shfl_down(m, off, 32));
    m = __shfl(m, 0, 32);
    const float e0 = __expf(l0 - m), e1 = __expf(l1 - m);
    float s = e0 + e1;
    for (int off = 16; off > 0; off >>= 1) s += __shfl_down(s, off, 32);
    s = __shfl(s, 0, 32);
    const float inv = 1.0f / s;
    atomicAdd(&s_psum[lane],      e0 * inv);
    atomicAdd(&s_psum[lane + 32], e1 * inv);

    __syncthreads();
    if (threadIdx.x < NE) {
        atomicAdd(&g_counts[threadIdx.x], s_counts[threadIdx.x]);
        atomicAdd(&g_psum[threadIdx.x],   s_psum[threadIdx.x]);
    }
}

// ---------------------------------------------------------------------------
// loss = NE * sum_e (counts[e]/B) * (psum[e]/B)
// ---------------------------------------------------------------------------
__global__ void router_finalize(const float* __restrict__ g_counts,
                                const float* __restrict__ g_psum,
                                float* __restrict__ loss) {
    if (threadIdx.x == 0) {
        float acc = 0.0f;
        const float invB = 1.0f / (float)BATCH;
        for (int e = 0; e < NE; ++e)
            acc += (g_counts[e] * invB) * (g_psum[e] * invB);
        *loss = (float)NE * acc;
    }
}

extern "C" void kernel_launch(void* const* d_in, const int* in_sizes, int n_in,
                              void* d_out, int out_size, void* d_ws, size_t ws_size,
                              hipStream_t stream) {
    (void)in_sizes; (void)n_in; (void)out_size; (void)ws_size;
    const float* x     = (const float*)d_in[0];
    const float* W     = (const float*)d_in[1];
    const float* noise = (const float*)d_in[2];

    float* out        = (float*)d_out;
    float* out_idx    = out;                     // 8192*2
    float* out_w      = out + BATCH * 2;         // 8192*2
    float* out_logits = out + BATCH * 4;         // 8192*64
    float* out_loss   = out + BATCH * 4 + BATCH * NE;

    float* ws      = (float*)d_ws;
    float* w_count = ws;        // [64]
    float* w_psum  = ws + NE;   // [64]

    router_init    <<<1,             128, 0, stream>>>(ws);
    router_gemm    <<<BATCH / 16,    128, 0, stream>>>(x, W, out_logits);
    router_rowwise <<<BATCH / 8,     256, 0, stream>>>(out_logits, noise,
                                                       out_idx, out_w,
                                                       w_count, w_psum);
    router_finalize<<<1,              64, 0, stream>>>(w_count, w_psum, out_loss);
}